// NavMapping_32263794327621
// MI455X (gfx1250) — compile-verified
//
#include <hip/hip_runtime.h>
#include <math.h>

// ---------------- constants from the reference ----------------
#define H_SCR   1024
#define W_SCR   1024
#define N_PIX   (H_SCR * W_SCR)
#define MAPC    960
#define N_MAP   (MAPC * MAPC)          // 921600 per channel
#define VRD     100
#define ZH      80                     // MAX_H - MIN_H
#define N_VOX   (VRD * VRD * ZH)       // 800000
#define MINZ    13                     // 25/5 - (-8)
#define MAXZ    25                     // (88+1)/5 - (-8)

// output layout (floats): [fp_map 10000][map_pred 2*921600][cur 3][cur 3]
#define OUT_MAP_OFF 10000
#define OUT_CUR_OFF (10000 + 2 * N_MAP)

// workspace layout (floats)
#define WS_VOX 0
#define WS_PAR 800000                  // {cos, sin, tx, ty}
#define WS_FP  800016                  // 2*10000 floats, 64B aligned for b128 async

typedef __attribute__((ext_vector_type(2))) float v2f;
typedef __attribute__((ext_vector_type(8))) float v8f;

// ---------------- kernel 1: zero the voxel accumulator ----------------
__global__ void __launch_bounds__(256) k_zero(float* __restrict__ vox) {
    int i = blockIdx.x * blockDim.x + threadIdx.x;
    if (i < N_VOX) vox[i] = 0.0f;
}

// ---------------- kernel 2: pose update (1 thread) ----------------
__global__ void k_pose(const float* __restrict__ pose_obs,
                       const float* __restrict__ poses_last,
                       float* __restrict__ par, float* __restrict__ out_cur) {
    if (threadIdx.x != 0 || blockIdx.x != 0) return;
    const float R2D = 57.29577951308232f;
    float th = poses_last[2] / R2D;
    float s = sinf(th), c = cosf(th);
    float ny = poses_last[1] + pose_obs[0] * s + pose_obs[1] * c;
    float nx = poses_last[0] + pose_obs[0] * c - pose_obs[1] * s;
    float nt = poses_last[2] + pose_obs[2] * R2D;
    nt = fmodf(nt - 180.0f, 360.0f) + 180.0f;
    nt = fmodf(nt + 180.0f, 360.0f) - 180.0f;
    out_cur[0] = nx; out_cur[1] = ny; out_cur[2] = nt;
    out_cur[3] = nx; out_cur[4] = ny; out_cur[5] = nt;
    // st = [sp_y, sp_x, nt-180]; theta1 rotation uses st[2], theta2 uses (st[0], st[1])
    float sp0 = -(nx * 20.0f - 480.0f) / 480.0f;   // from x
    float sp1 = -(ny * 20.0f - 480.0f) / 480.0f;   // from y
    float t = (nt - 180.0f) * 0.017453292519943295f;
    par[0] = cosf(t);
    par[1] = sinf(t);
    par[2] = sp1;   // translate grid x
    par[3] = sp0;   // translate grid y
}

// ---------------- kernel 3: point cloud + trilinear scatter splat ----------------
__global__ void __launch_bounds__(256) k_splat(const float* __restrict__ depth,
                                               const float* __restrict__ feat,
                                               const int* __restrict__ elev_p,
                                               float* __restrict__ vox) {
    int n = blockIdx.x * blockDim.x + threadIdx.x;
    if (n >= N_PIX) return;
    int row = n / W_SCR, col = n - row * W_SCR;
    float d = depth[n];
    const float ctr = 511.5f, f = 512.0f;           // f = 512/tan(45deg)
    float X = ((float)col - ctr) * d / f;
    float Z = ((float)(H_SCR - 1 - row) - ctr) * d / f;
    float Y = d;
    // Rx(elev)
    float a = (float)(*elev_p) * 0.017453292519943295f;
    float ca = cosf(a), sa = sinf(a);
    float y1 = ca * Y - sa * Z;
    float z1 = sa * Y + ca * Z + 88.0f;             // + AGENT_H
    // Rz(-pi/2): cos = 6.123234e-17 (float32 of np.cos(-pi/2)), sin = -1
    const float cb = 6.123234e-17f;
    float x2 = cb * X + y1;
    float y2 = -X + cb * y1;
    x2 += 250.0f;                                   // SHIFT_X
    // normalized coords -> splat positions (mirror reference arithmetic)
    float cx = (x2 / 5.0f - 50.0f) / 100.0f * 2.0f;
    float cy = (y2 / 5.0f - 50.0f) / 100.0f * 2.0f;
    float cz = (z1 / 5.0f - 32.0f) / 80.0f * 2.0f;
    float pos0 = cx * 50.0f + 50.0f;
    float pos1 = cy * 50.0f + 50.0f;
    float pos2 = cz * 40.0f + 40.0f;

    float p0[2], w0[2], p1[2], w1[2], p2[2], w2[2];
    float fl;
    fl = floorf(pos0);
#pragma unroll
    for (int o = 0; o < 2; ++o) {
        float pp = fl + (float)o;
        bool safe = (pp > 0.0f) && (pp < 100.0f);
        w0[o] = safe ? (1.0f - fabsf(pos0 - pp)) : 0.0f;
        p0[o] = safe ? pp : 0.0f;
    }
    fl = floorf(pos1);
#pragma unroll
    for (int o = 0; o < 2; ++o) {
        float pp = fl + (float)o;
        bool safe = (pp > 0.0f) && (pp < 100.0f);
        w1[o] = safe ? (1.0f - fabsf(pos1 - pp)) : 0.0f;
        p1[o] = safe ? pp : 0.0f;
    }
    fl = floorf(pos2);
#pragma unroll
    for (int o = 0; o < 2; ++o) {
        float pp = fl + (float)o;
        bool safe = (pp > 0.0f) && (pp < 80.0f);
        w2[o] = safe ? (1.0f - fabsf(pos2 - pp)) : 0.0f;
        p2[o] = safe ? pp : 0.0f;
    }
    float fv = feat[n];
#pragma unroll
    for (int c0 = 0; c0 < 2; ++c0)
#pragma unroll
        for (int c1 = 0; c1 < 2; ++c1)
#pragma unroll
            for (int c2 = 0; c2 < 2; ++c2) {
                float w = w0[c0] * w1[c1] * w2[c2] * fv;
                if (w != 0.0f) {
                    int idx = ((int)p0[c0] * 100 + (int)p1[c1]) * 80 + (int)p2[c2];
                    atomicAdd(&vox[idx], w);
                }
            }
}

// ---------------- kernel 4: per-voxel round + z-reduction via WMMA ----------------
// Column sums over 80 z-values == (16xK A) x (ones B) matvec, chained over K=4 steps.
// A layout for V_WMMA_F32_16X16X4_F32: lane m (<16) holds A[m][0..1], lane m+16 holds
// A[m][2..3]. B = all-ones, so its lane layout is irrelevant. Row sums appear at N=0:
// lanes 0 (rows 0-7) and 16 (rows 8-15) of the v8f accumulator.
__global__ void __launch_bounds__(256) k_reduce(const float* __restrict__ vox,
                                                float* __restrict__ fp,
                                                float* __restrict__ out_fpmap) {
    int tid  = blockIdx.x * blockDim.x + threadIdx.x;
#if defined(__gfx1250__) && __has_builtin(__builtin_amdgcn_wmma_f32_16x16x4_f32)
    int wave = tid >> 5;
    int lane = threadIdx.x & 31;
    if (wave >= 625) return;          // uniform per wave -> EXEC all-ones inside
    int m = lane & 15, hi = lane >> 4;
    int colb = wave * 16;
    int col  = colb + m;
    v8f cexp = {0.f, 0.f, 0.f, 0.f, 0.f, 0.f, 0.f, 0.f};
    v8f cmap = {0.f, 0.f, 0.f, 0.f, 0.f, 0.f, 0.f, 0.f};
    v2f b = {1.0f, 1.0f};
    for (int s = 0; s < 20; ++s) {
        int k0 = s * 4 + hi * 2;
        float v0 = rintf(vox[col * 80 + k0]);
        float v1 = rintf(vox[col * 80 + k0 + 1]);
        v2f a  = {v0, v1};
        v2f am = {(k0     >= MINZ && k0     < MAXZ) ? v0 : 0.0f,
                  (k0 + 1 >= MINZ && k0 + 1 < MAXZ) ? v1 : 0.0f};
        cexp = __builtin_amdgcn_wmma_f32_16x16x4_f32(false, a,  false, b, (short)0, cexp, false, false);
        cmap = __builtin_amdgcn_wmma_f32_16x16x4_f32(false, am, false, b, (short)0, cmap, false, false);
    }
    if (m == 0) {
#pragma unroll
        for (int i = 0; i < 8; ++i) {
            int r    = hi * 8 + i;
            int colr = colb + r;
            int d0 = colr / 100, d1 = colr - d0 * 100;
            int off = (99 - d1) * 100 + d0;     // flip + swapaxes
            float mv = fminf(fmaxf(cmap[i], 0.0f), 1.0f);
            float ev = fminf(fmaxf(cexp[i], 0.0f), 1.0f);
            fp[off]          = mv;
            fp[10000 + off]  = ev;
            out_fpmap[off]   = mv;
        }
    }
#else
    if (tid >= VRD * VRD) return;
    int d0 = tid / 100, d1 = tid - d0 * 100;
    float se = 0.0f, sm = 0.0f;
    for (int z = 0; z < ZH; ++z) {
        float v = rintf(vox[tid * 80 + z]);
        se += v;
        if (z >= MINZ && z < MAXZ) sm += v;
    }
    int off = (99 - d1) * 100 + d0;
    float mv = fminf(fmaxf(sm, 0.0f), 1.0f);
    float ev = fminf(fmaxf(se, 0.0f), 1.0f);
    fp[off]         = mv;
    fp[10000 + off] = ev;
    out_fpmap[off]  = mv;
#endif
}

// ---------------- kernel 5: fused double grid-sample + max with maps_last ----------
// Stage the 2x100x100 nonzero window (80 KB) into LDS with CDNA5 async global->LDS
// copies (bypasses VGPRs, tracked by ASYNCcnt), then each output pixel does its 16
// taps against LDS. av is zero outside the window, so no other map data is needed.
__global__ void __launch_bounds__(256) k_sample(const float* __restrict__ fp,
                                                const float* __restrict__ par,
                                                const float* __restrict__ maps_last,
                                                float* __restrict__ out_map) {
    extern __shared__ float lds_fp[];   // 20000 floats
    {
        unsigned lbase = (unsigned)(size_t)(void*)lds_fp;   // low 32 bits = LDS offset
        unsigned long long gbase = (unsigned long long)(size_t)fp;
        for (int i = threadIdx.x; i < 5000; i += blockDim.x) {
            unsigned loff = lbase + (unsigned)i * 16u;
            unsigned long long g = gbase + (unsigned long long)i * 16ull;
            asm volatile("global_load_async_to_lds_b128 %0, %1, off"
                         :: "v"(loff), "v"(g) : "memory");
        }
        asm volatile("s_wait_asynccnt 0" ::: "memory");
    }
    __syncthreads();

    int pix = blockIdx.x * blockDim.x + threadIdx.x;
    if (pix >= N_MAP) return;
    int h = pix / MAPC, w = pix - h * MAPC;
    float cr = par[0], sr = par[1], tx = par[2], ty = par[3];

    float gx = ((float)(2 * w + 1)) / 960.0f - 1.0f;
    float gy = ((float)(2 * h + 1)) / 960.0f - 1.0f;
    // outer (translation) sample coords into the rotated image
    float xs = (gx + tx + 1.0f) * 0.5f * 959.0f;
    float ys = (gy + ty + 1.0f) * 0.5f * 959.0f;
    float x0 = floorf(xs), y0 = floorf(ys);
    float wx = xs - x0, wy = ys - y0;
    float acc0 = 0.0f, acc1 = 0.0f;
#pragma unroll
    for (int dy = 0; dy < 2; ++dy)
#pragma unroll
        for (int dx = 0; dx < 2; ++dx) {
            float fx = x0 + (float)dx, fy = y0 + (float)dy;
            float wgt = (dx ? wx : 1.0f - wx) * (dy ? wy : 1.0f - wy);
            if (fx < 0.0f || fx >= 960.0f || fy < 0.0f || fy >= 960.0f || wgt == 0.0f)
                continue;
            int ii = (int)fx, jj = (int)fy;
            // inner (rotation) sample of av at integer pixel (jj, ii)
            float gx2 = ((float)(2 * ii + 1)) / 960.0f - 1.0f;
            float gy2 = ((float)(2 * jj + 1)) / 960.0f - 1.0f;
            float rx = cr * gx2 - sr * gy2;
            float ry = sr * gx2 + cr * gy2;
            float px = (rx + 1.0f) * 0.5f * 959.0f;
            float py = (ry + 1.0f) * 0.5f * 959.0f;
            float qx = floorf(px), qy = floorf(py);
            float ax = px - qx, ay = py - qy;
            float v0 = 0.0f, v1 = 0.0f;
#pragma unroll
            for (int ey = 0; ey < 2; ++ey)
#pragma unroll
                for (int ex = 0; ex < 2; ++ex) {
                    float cxp = qx + (float)ex, cyp = qy + (float)ey;
                    if (cxp < 0.0f || cxp >= 960.0f || cyp < 0.0f || cyp >= 960.0f)
                        continue;
                    int xi = (int)cxp, yi = (int)cyp;
                    if (xi >= 430 && xi < 530 && yi >= 380 && yi < 480) {
                        float wav = (ex ? ax : 1.0f - ax) * (ey ? ay : 1.0f - ay);
                        int o = (yi - 380) * 100 + (xi - 430);
                        v0 += wav * lds_fp[o];
                        v1 += wav * lds_fp[10000 + o];
                    }
                }
            acc0 += wgt * v0;
            acc1 += wgt * v1;
        }
    out_map[pix]         = fmaxf(maps_last[pix], acc0);
    out_map[N_MAP + pix] = fmaxf(maps_last[N_MAP + pix], acc1);
}

// ---------------- launch ----------------
extern "C" void kernel_launch(void* const* d_in, const int* in_sizes, int n_in,
                              void* d_out, int out_size, void* d_ws, size_t ws_size,
                              hipStream_t stream) {
    const float* depth      = (const float*)d_in[0];
    const float* pose_obs   = (const float*)d_in[1];
    const float* maps_last  = (const float*)d_in[2];
    const float* poses_last = (const float*)d_in[3];
    const float* feat       = (const float*)d_in[4];
    const int*   elev       = (const int*)d_in[5];
    float* out = (float*)d_out;
    float* ws  = (float*)d_ws;
    float* vox = ws + WS_VOX;
    float* par = ws + WS_PAR;
    float* fp  = ws + WS_FP;

    k_zero  <<<(N_VOX + 255) / 256, 256, 0, stream>>>(vox);
    k_pose  <<<1, 32, 0, stream>>>(pose_obs, poses_last, par, out + OUT_CUR_OFF);
    k_splat <<<(N_PIX + 255) / 256, 256, 0, stream>>>(depth, feat, elev, vox);
    k_reduce<<<79, 256, 0, stream>>>(vox, fp, out);
    k_sample<<<(N_MAP + 255) / 256, 256, 20000 * sizeof(float), stream>>>(
        fp, par, maps_last, out + OUT_MAP_OFF);
}